// TripletLoss_84997402788383
// MI455X (gfx1250) — compile-verified
//
#include <hip/hip_runtime.h>
#include <math.h>

// ---------------------------------------------------------------------------
// TripletLoss (batch-hard) for MI455X / gfx1250.
//
// Fused G = E*E^T via V_WMMA_F32_16X16X4_F32 folded straight into per-row
// hardest-pos / hardest-neg accumulators (64MB distance matrix never touches
// HBM; E is 4MB and L2-resident at 192MB).
//
//   * 1x4 column tiling per wave: one A fragment feeds 4 independent WMMA
//     accumulator chains (less DS bandwidth per FLOP, more XDL ILP).
//   * Tensor Data Mover staging: A block + double-buffered B tiles are moved
//     by TENSOR_LOAD_TO_LDS with descriptor padding (pad_interval=256 dwords,
//     pad_amount=4 dwords) so the DMA itself writes the conflict-free
//     260-float LDS row stride. Overlap is via TENSORcnt + split barriers.
//   * This toolchain exposes the 6-arg tensor_load_to_lds builtin
//     (g0:u32x4, g1:i32x8, g2:i32x4, g3:i32x4, extra:i32x8, cpol).
// ---------------------------------------------------------------------------

typedef float v2f  __attribute__((ext_vector_type(2)));
typedef float v8f  __attribute__((ext_vector_type(8)));
typedef unsigned int u32x4 __attribute__((ext_vector_type(4)));
typedef int   i32x4 __attribute__((ext_vector_type(4)));
typedef int   i32x8 __attribute__((ext_vector_type(8)));

#define B_N            4096
#define D_K            256
#define LDS_STRIDE     260      // 256 + 4-float pad -> conflict-free DS reads
#define ROWS_PER_BLOCK 128      // 8 waves x 16-row tiles
#define COLS_PER_STAGE 64       // 4 x 16-column tiles per wave per stage
#define NSTAGES        (B_N / COLS_PER_STAGE)
#define MARGIN         0.3f
#define NEG_FILL       1e9f

#if defined(__has_builtin)
#if __has_builtin(__builtin_amdgcn_tensor_load_to_lds) && \
    __has_builtin(__builtin_amdgcn_s_wait_tensorcnt)
#define USE_TDM 1
#endif
#endif
#ifndef USE_TDM
#define USE_TDM 0
#endif

// ---------------------------------------------------------------------------
// Kernel 0: squared L2 norm of each embedding row. One wave32 per row.
// ---------------------------------------------------------------------------
__global__ void rownorm_kernel(const float* __restrict__ E,
                               float* __restrict__ sq) {
  const int lane = threadIdx.x & 31;
  const int row  = blockIdx.x * (blockDim.x >> 5) + (threadIdx.x >> 5);
  const float4* e4 = (const float4*)(E + (size_t)row * D_K);
  float4 a = e4[lane];
  float4 b = e4[lane + 32];
  float s = a.x * a.x + a.y * a.y + a.z * a.z + a.w * a.w +
            b.x * b.x + b.y * b.y + b.z * b.z + b.w * b.w;
#pragma unroll
  for (int off = 16; off > 0; off >>= 1) s += __shfl_xor(s, off, 32);
  if (lane == 0) sq[row] = s;
}

#if USE_TDM
// ---------------------------------------------------------------------------
// Issue one TDM 2D tile load: rows x 256 f32, row-major global -> LDS with a
// 4-dword pad after every 256 dwords (LDS row stride = LDS_STRIDE floats).
// D# layout per CDNA5 ISA (8.3 / 8.4).
// ---------------------------------------------------------------------------
__device__ __forceinline__ void tdm_load_rows(const float* gaddr,
                                              unsigned lds_byte_off,
                                              int rows) {
  const unsigned long long ga = (unsigned long long)(uintptr_t)gaddr;
  u32x4 g0;
  g0[0] = 1u;                                   // count=1, user descriptor
  g0[1] = lds_byte_off;                         // lds_addr [63:32]
  g0[2] = (unsigned)(ga & 0xFFFFFFFFu);         // global_addr[31:0]
  g0[3] = (unsigned)((ga >> 32) & 0x1FFFFFFu)   // global_addr[56:32]
        | (2u << 30);                           // type=2 (image)
  i32x8 g1;
  g1[0] = (2 << 16)                             // data_size = 4B
        | (1 << 20)                             // pad_enable
        | (7 << 22)                             // pad_interval: 256 dwords
        | (3 << 25);                            // pad_amount:   4 dwords
  g1[1] = (int)((unsigned)(D_K & 0xFFFF) << 16);   // tensor_dim0[15:0] @63:48
  g1[2] = (int)((unsigned)(B_N & 0xFFFF) << 16);   // tensor_dim1[15:0] @95:80
  g1[3] = (int)((unsigned)(D_K & 0xFFFF) << 16);   // tile_dim0 @127:112
  g1[4] = rows & 0xFFFF;                           // tile_dim1 @143:128
  g1[5] = D_K;                                     // tensor_dim0_stride @191:160
  g1[6] = 0;
  g1[7] = 0;
  i32x4 z4 = {};
  i32x8 z8 = {};
  __builtin_amdgcn_tensor_load_to_lds(g0, g1, z4, z4, z8, 0);
}

__device__ __forceinline__ unsigned lds_offset_of(const void* p) {
  // Flat shared addresses: [63:32] = shared aperture, [31:0] = LDS byte addr.
  return (unsigned)(uintptr_t)p;
}
#endif

// ---------------------------------------------------------------------------
// Kernel 1: fused Gram-tile (f32 WMMA) + batch-hard mining.
// 256 threads = 8 waves; wave w owns anchor rows [rowBlock+16w, +16).
// Per stage: 64 columns = 4 x (16x16) tiles per wave, 4 independent
// accumulator chains sharing each A fragment.
// ---------------------------------------------------------------------------
__global__ void hardest_kernel(const float* __restrict__ E,
                               const int*  __restrict__ labels,
                               const float* __restrict__ sq,
                               float* __restrict__ hpos,
                               float* __restrict__ hneg) {
  extern __shared__ float lds[];
  float* A_lds = lds;                                   // [128][LDS_STRIDE]
  float* B0    = lds + ROWS_PER_BLOCK * LDS_STRIDE;     // [ 64][LDS_STRIDE]
  float* B1    = B0 + COLS_PER_STAGE * LDS_STRIDE;      // [ 64][LDS_STRIDE]

  const int tid      = threadIdx.x;
  const int lane     = tid & 31;
  const int wave     = tid >> 5;
  const int rowBlock = blockIdx.x * ROWS_PER_BLOCK;

#if USE_TDM
  // Prologue DMAs: A block + B stage 0 (in-order on TENSORcnt).
  if (wave == 0) {
    tdm_load_rows(E + (size_t)rowBlock * D_K, lds_offset_of(A_lds),
                  ROWS_PER_BLOCK);
    tdm_load_rows(E, lds_offset_of(B0), COLS_PER_STAGE);
  }
#else
  // Fallback: cooperative vectorized staging of the A block.
  for (int i = tid; i < ROWS_PER_BLOCK * (D_K / 4); i += blockDim.x) {
    const int r  = i >> 6;
    const int c4 = i & 63;
    float4 v = ((const float4*)(E + (size_t)(rowBlock + r) * D_K))[c4];
    float* dst = &A_lds[r * LDS_STRIDE + c4 * 4];
    dst[0] = v.x; dst[1] = v.y; dst[2] = v.z; dst[3] = v.w;
  }
#endif

  // Per-lane anchor metadata (C layout: lane holds rows v + 8*half, one col).
  const int half      = lane >> 4;
  const int n_in_tile = lane & 15;
  const int myRowBase = rowBlock + wave * 16 + half * 8;
  float sqr[8];
  int   lr[8];
#pragma unroll
  for (int v = 0; v < 8; ++v) {
    sqr[v] = sq[myRowBase + v];
    lr[v]  = labels[myRowBase + v];
  }

  float posmax[8], negmin[8];
#pragma unroll
  for (int v = 0; v < 8; ++v) { posmax[v] = -1.0f; negmin[v] = NEG_FILL; }

  // WMMA f32 16x16x4 fragment addressing (ISA 7.12.2):
  //   lane m/n = lane&15, VGPR0/1 hold K = 2*(lane>>4) + {0,1}
  const int klane = half * 2;
  const float* Arow = &A_lds[(wave * 16 + n_in_tile) * LDS_STRIDE + klane];

  for (int ct = 0; ct < NSTAGES; ++ct) {
    const int colBase = ct * COLS_PER_STAGE;

#if USE_TDM
    if (wave == 0) {
      if (ct + 1 < NSTAGES) {
        // Issue next stage into the other buffer (readers of that buffer
        // finished before last iteration's trailing barrier).
        tdm_load_rows(E + (size_t)(colBase + COLS_PER_STAGE) * D_K,
                      lds_offset_of((ct & 1) ? B0 : B1), COLS_PER_STAGE);
        __builtin_amdgcn_s_wait_tensorcnt(1);   // current stage complete
      } else {
        __builtin_amdgcn_s_wait_tensorcnt(0);
      }
    }
    __syncthreads();
    const float* Bcur = (ct & 1) ? B1 : B0;
#else
    const float* Bcur = B0;
    for (int i = tid; i < COLS_PER_STAGE * (D_K / 4); i += blockDim.x) {
      const int r  = i >> 6;
      const int c4 = i & 63;
      float4 v = ((const float4*)(E + (size_t)(colBase + r) * D_K))[c4];
      float* dst = &B0[r * LDS_STRIDE + c4 * 4];
      dst[0] = v.x; dst[1] = v.y; dst[2] = v.z; dst[3] = v.w;
    }
    __syncthreads();
#endif

    // ---- 4 x (16x16) Gram tiles, K=256: 256 WMMAs, 4 independent chains ---
    v8f acc[4] = {};
    const float* Brow = Bcur + n_in_tile * LDS_STRIDE + klane;
#pragma unroll
    for (int kt = 0; kt < D_K / 4; ++kt) {
      v2f a = *(const v2f*)(Arow + kt * 4);
#pragma unroll
      for (int t = 0; t < 4; ++t) {
        v2f b = *(const v2f*)(Brow + t * 16 * LDS_STRIDE + kt * 4);
        acc[t] = __builtin_amdgcn_wmma_f32_16x16x4_f32(
            /*neg_a=*/false, a, /*neg_b=*/false, b,
            /*c_mod=*/(short)0, acc[t], /*reuse_a=*/false, /*reuse_b=*/false);
      }
    }

    // ---- fused epilogue: distance + masks + running max/min ---------------
#pragma unroll
    for (int t = 0; t < 4; ++t) {
      const int   col = colBase + t * 16 + n_in_tile;  // one col per lane
      const float sqc = sq[col];
      const int   lc  = labels[col];
#pragma unroll
      for (int v = 0; v < 8; ++v) {
        float d2 = fmaxf(sqr[v] + sqc - 2.0f * acc[t][v], 0.0f);
        float d  = (d2 > 0.0f) ? sqrtf(d2) : 0.0f;   // reference zero guard
        const bool same = (lr[v] == lc);
        const bool diag = ((myRowBase + v) == col);
        if (same && !diag) posmax[v] = fmaxf(posmax[v], d);
        if (!same)         negmin[v] = fminf(negmin[v], d);
      }
    }
    __syncthreads();   // readers done before this buffer is re-written
  }

  // ---- reduce each row over its 16 lanes (xor masks stay in the half) -----
#pragma unroll
  for (int v = 0; v < 8; ++v) {
#pragma unroll
    for (int off = 1; off < 16; off <<= 1) {
      posmax[v] = fmaxf(posmax[v], __shfl_xor(posmax[v], off, 32));
      negmin[v] = fminf(negmin[v], __shfl_xor(negmin[v], off, 32));
    }
  }
  if (n_in_tile == 0) {
#pragma unroll
    for (int v = 0; v < 8; ++v) {
      hpos[myRowBase + v] = posmax[v];
      hneg[myRowBase + v] = negmin[v];
    }
  }
}

// ---------------------------------------------------------------------------
// Kernel 2: hinge + valid-mean over the 4096 anchors -> scalar loss.
// ---------------------------------------------------------------------------
__global__ void loss_kernel(const float* __restrict__ hpos,
                            const float* __restrict__ hneg,
                            float* __restrict__ out) {
  __shared__ float s_tot[256];
  __shared__ float s_cnt[256];
  const int tid = threadIdx.x;
  float tot = 0.0f, cnt = 0.0f;
  for (int r = tid; r < B_N; r += 256) {
    const float hp = hpos[r];
    const float hn = hneg[r];
    if (hp >= 0.0f && hn < NEG_FILL) {   // has_pos && has_neg
      tot += fmaxf(hp - hn + MARGIN, 0.0f);
      cnt += 1.0f;
    }
  }
  s_tot[tid] = tot; s_cnt[tid] = cnt;
  __syncthreads();
  for (int s = 128; s > 0; s >>= 1) {
    if (tid < s) { s_tot[tid] += s_tot[tid + s]; s_cnt[tid] += s_cnt[tid + s]; }
    __syncthreads();
  }
  if (tid == 0)
    out[0] = (s_cnt[0] > 0.0f) ? s_tot[0] / fmaxf(s_cnt[0], 1.0f) : 0.0f;
}

// ---------------------------------------------------------------------------
extern "C" void kernel_launch(void* const* d_in, const int* in_sizes, int n_in,
                              void* d_out, int out_size, void* d_ws,
                              size_t ws_size, hipStream_t stream) {
  const float* E      = (const float*)d_in[0];   // [4096, 256] fp32
  const int*   labels = (const int*)d_in[1];     // [4096]
  float* out = (float*)d_out;

  // workspace: sq[4096] | hpos[4096] | hneg[4096]
  float* sq   = (float*)d_ws;
  float* hpos = sq + B_N;
  float* hneg = hpos + B_N;

  rownorm_kernel<<<B_N / 8, 256, 0, stream>>>(E, sq);

  // LDS: A(128x260) + 2 x B(64x260) floats = 266,240 B (< 320KB/WGP).
  const size_t ldsBytes =
      (size_t)(ROWS_PER_BLOCK + 2 * COLS_PER_STAGE) * LDS_STRIDE * sizeof(float);
  hardest_kernel<<<B_N / ROWS_PER_BLOCK, 256, ldsBytes, stream>>>(
      E, labels, sq, hpos, hneg);

  loss_kernel<<<1, 256, 0, stream>>>(hpos, hneg, out);
}